// Smoother_2808908612179
// MI455X (gfx1250) — compile-verified
//
#include <hip/hip_runtime.h>
#include <math.h>

// Problem constants (from the reference)
#define NB   8
#define LSEQ 4096
#define CH   256
#define NW   4088          // LSEQ - WIN + 1
#define WIN  9

typedef __attribute__((ext_vector_type(2))) float v2f;
typedef __attribute__((ext_vector_type(8))) float v8f;

// ---------------------------------------------------------------------------
// Kernel 1: Q = x @ W_qry, K = x @ W_key   (both (NB*LSEQ) x CH, fp32 WMMA)
//
// One wave computes one 16x16 output tile. Global wave id encodes
// (rowTile 0..2047, colTile 0..31); colTile<16 -> Q, else -> K.
// A-frag (16x4 f32): lane = m + 16*(k>>1), vgpr = k&1  -> contiguous float2/lane
// B-frag (4x16 f32): lane = n + 16*(k>>1), vgpr = k&1
// C/D  (16x16 f32):  vgpr g -> row g + 8*(lane>>4), col = lane&15
// ---------------------------------------------------------------------------
__global__ void __launch_bounds__(256)
gemm_qk_wmma(const float* __restrict__ x,
             const float* __restrict__ Wq,
             const float* __restrict__ Wk,
             float* __restrict__ Qbuf,
             float* __restrict__ Kbuf)
{
    const int lane = threadIdx.x & 31;
    const int wave = threadIdx.x >> 5;
    const int gwid = blockIdx.x * 8 + wave;      // 0 .. 65535
    const int ct   = gwid & 31;                  // 32 col tiles (16 Q + 16 K)
    const int rt   = gwid >> 5;                  // 2048 row tiles
    const int row0 = rt * 16;
    const int col0 = (ct & 15) * 16;

    const float* __restrict__ W   = (ct >= 16) ? Wk   : Wq;
    float*       __restrict__ Out = (ct >= 16) ? Kbuf : Qbuf;

    const int half = lane >> 4;                  // selects K pair {0,1} vs {2,3}
    const int mn   = lane & 15;                  // row (A) / col (B) within tile

    const float* __restrict__ arow = x + (size_t)(row0 + mn) * CH;
    const float* __restrict__ bcol = W + col0 + mn;

    v8f acc = {};
#pragma unroll 4
    for (int kk = 0; kk < CH; kk += 4) {
        const int ka = kk + 2 * half;
        // A: two consecutive k values of this lane's row -> single b64 load
        v2f a = *reinterpret_cast<const v2f*>(arow + ka);
        // B: W[ka][col], W[ka+1][col]
        v2f bf;
        bf.x = bcol[(size_t)ka * CH];
        bf.y = bcol[(size_t)(ka + 1) * CH];
        acc = __builtin_amdgcn_wmma_f32_16x16x4_f32(
                  /*neg_a=*/false, a, /*neg_b=*/false, bf,
                  /*c_mod=*/(short)0, acc, /*reuse_a=*/false, /*reuse_b=*/false);
    }

    float* __restrict__ orow = Out + (size_t)row0 * CH + col0 + mn;
#pragma unroll
    for (int g = 0; g < 8; ++g)
        orow[(size_t)(g + 8 * half) * CH] = acc[g];
}

// ---------------------------------------------------------------------------
// Kernel 2: fused score / softmax / weighted-combine.
// Block = 256 threads (8 waves) handles TW=16 consecutive window positions
// of one batch item. Stages the shared x / K rows [w0, w0+TW+8) in LDS
// (24 rows x 256 f32 x 2 arrays = 48 KB). Each wave owns 2 window positions;
// within a wave each lane owns 8 of the 256 u (and later c) channels.
// ---------------------------------------------------------------------------
#define TW   16
#define ROWS (TW + WIN - 1)   // 24

__global__ void __launch_bounds__(256)
attn_smooth(const float* __restrict__ x,
            const float* __restrict__ Qbuf,
            const float* __restrict__ Kbuf,
            const float* __restrict__ vvec,
            const float* __restrict__ bvec,
            float* __restrict__ out)
{
    __shared__ float sx[ROWS * CH];
    __shared__ float sk[ROWS * CH];
    __shared__ float sb[CH];
    __shared__ float sv[CH];

    const int n   = blockIdx.y;
    const int w0  = blockIdx.x * TW;
    const int tid = threadIdx.x;

    // Cooperative staging: ROWS*CH = 6144 floats = 1536 float4 per array.
    for (int i = tid; i < ROWS * CH / 4; i += 256) {
        const int r  = i >> 6;                 // 64 float4 per row
        const int c4 = i & 63;
        int gr = w0 + r;
        if (gr > LSEQ - 1) gr = LSEQ - 1;      // clamp tail (unused rows)
        const size_t gbase = ((size_t)n * LSEQ + gr) * CH;
        reinterpret_cast<float4*>(sx)[i] =
            reinterpret_cast<const float4*>(x + gbase)[c4];
        reinterpret_cast<float4*>(sk)[i] =
            reinterpret_cast<const float4*>(Kbuf + gbase)[c4];
    }
    sb[tid] = bvec[tid];
    sv[tid] = vvec[tid];
    __syncthreads();

    const int wave = tid >> 5;
    const int lane = tid & 31;
    const int u0   = lane * 8;

    float bb[8], vv[8];
#pragma unroll
    for (int i = 0; i < 8; ++i) { bb[i] = sb[u0 + i]; vv[i] = sv[u0 + i]; }

    for (int j = 0; j < 2; ++j) {
        const int wl = wave * 2 + j;           // 0..15 local window index
        const int w  = w0 + wl;
        if (w >= NW) continue;

        // Center query row (w+4), this lane's 8 channels.
        const float* qp = Qbuf + ((size_t)n * LSEQ + (w + WIN / 2)) * CH + u0;
        float q[8];
        {
            float4 q0 = *reinterpret_cast<const float4*>(qp);
            float4 q1 = *reinterpret_cast<const float4*>(qp + 4);
            q[0]=q0.x; q[1]=q0.y; q[2]=q0.z; q[3]=q0.w;
            q[4]=q1.x; q[5]=q1.y; q[6]=q1.z; q[7]=q1.w;
        }

        // Scores over the 9 window slots, reduced across the wave.
        float sc[WIN];
#pragma unroll
        for (int s = 0; s < WIN; ++s) {
            const float* kp = sk + (wl + s) * CH + u0;
            float p = 0.0f;
#pragma unroll
            for (int i = 0; i < 8; ++i)
                p += tanhf(q[i] + kp[i] + bb[i]) * vv[i];
            // wave32 all-reduce (every lane ends with the full sum)
#pragma unroll
            for (int off = 16; off > 0; off >>= 1)
                p += __shfl_xor(p, off, 32);
            sc[s] = p;
        }

        // 9-way softmax, replicated in every lane.
        float mx = sc[0];
#pragma unroll
        for (int s = 1; s < WIN; ++s) mx = fmaxf(mx, sc[s]);
        float den = 0.0f, wt[WIN];
#pragma unroll
        for (int s = 0; s < WIN; ++s) { wt[s] = __expf(sc[s] - mx); den += wt[s]; }
        const float inv = 1.0f / den;

        // Weighted combine of the 9 x-rows; lane owns channels u0..u0+7.
        float acc[8] = {0,0,0,0,0,0,0,0};
#pragma unroll
        for (int s = 0; s < WIN; ++s) {
            const float ws = wt[s] * inv;
            const float* xp = sx + (wl + s) * CH + u0;
#pragma unroll
            for (int i = 0; i < 8; ++i)
                acc[i] = fmaf(xp[i], ws, acc[i]);
        }

        float* op = out + ((size_t)n * NW + w) * CH + u0;
        float4 o0 = { acc[0], acc[1], acc[2], acc[3] };
        float4 o1 = { acc[4], acc[5], acc[6], acc[7] };
        *reinterpret_cast<float4*>(op)     = o0;
        *reinterpret_cast<float4*>(op + 4) = o1;
    }
}

// ---------------------------------------------------------------------------
extern "C" void kernel_launch(void* const* d_in, const int* in_sizes, int n_in,
                              void* d_out, int out_size, void* d_ws, size_t ws_size,
                              hipStream_t stream)
{
    const float* x  = (const float*)d_in[0];
    const float* Wq = (const float*)d_in[1];
    const float* Wk = (const float*)d_in[2];
    const float* v  = (const float*)d_in[3];
    const float* b  = (const float*)d_in[4];
    float* out = (float*)d_out;

    float* Qbuf = (float*)d_ws;                        // (NB*LSEQ, CH) fp32
    float* Kbuf = Qbuf + (size_t)NB * LSEQ * CH;       // (NB*LSEQ, CH) fp32

    // 2048 row tiles * 32 col tiles = 65536 waves / 8 waves per block.
    gemm_qk_wmma<<<8192, 256, 0, stream>>>(x, Wq, Wk, Qbuf, Kbuf);

    // ceil(NW / TW) = 256 blocks per batch item.
    attn_smooth<<<dim3((NW + TW - 1) / TW, NB), 256, 0, stream>>>(
        x, Qbuf, Kbuf, v, b, out);
}